// CausalSelfAttention_73426760893020
// MI455X (gfx1250) — compile-verified
//
#include <hip/hip_runtime.h>

// ---------------------------------------------------------------------------
// CausalSelfAttention forward for MI455X (gfx1250), f32 end-to-end.
//   1) gemm_wmma_f32<1>: qkv = x @ w_attn^T (4096x6144x2048); A/B tiles staged
//      via GLOBAL_LOAD_ASYNC_TO_LDS_B128 with double-buffered LDS (copy/compute
//      overlap on ASYNCcnt); fused head-index RoPE; scatter q/k/v -> (B,H,L,D).
//   2) flash_attn: streaming-softmax causal attention; K/V tiles DMA'd by the
//      Tensor Data Mover (tensor_load_to_lds, TENSORcnt) with hardware LDS
//      padding producing a conflict-free 132-float row stride; QK^T and PV on
//      V_WMMA_F32_16X16X4_F32 (f32 matrix pipe, matches f32 reference).
//   3) gemm_wmma_f32<0>: out = y @ w_proj^T (4096x2048x2048).
// Workspace: 3x (B*H*L*D) + (B*L*C) floats = 128 MB.
// ---------------------------------------------------------------------------

typedef __attribute__((ext_vector_type(2))) float    v2f;
typedef __attribute__((ext_vector_type(4))) float    v4f;
typedef __attribute__((ext_vector_type(8))) float    v8f;
typedef __attribute__((ext_vector_type(4))) unsigned v4u;
typedef __attribute__((ext_vector_type(8))) unsigned v8u;

#define B_   2
#define L_   2048
#define C_   2048
#define H_   16
#define D_   128
#define NQKV 6144

__device__ __forceinline__ v8f wmma4(v2f a, v2f b, v8f c) {
  // D = A(16x4,f32) x B(4x16,f32) + C(16x16,f32); wave32, 8-arg form.
  return __builtin_amdgcn_wmma_f32_16x16x4_f32(false, a, false, b, (short)0, c,
                                               false, false);
}

__device__ __forceinline__ v8f v8zero() {
  v8f z;
#pragma unroll
  for (int i = 0; i < 8; ++i) z[i] = 0.0f;
  return z;
}

// LDS byte offset of a __shared__ object: hardware truncates flat LDS
// addresses to addr[31:0], which is the workgroup-relative byte offset.
__device__ __forceinline__ unsigned lds_off(const void* p) {
  return (unsigned)(unsigned long long)p;
}

// Async memory->LDS copy, 16 bytes per enabled lane, tracked by ASYNCcnt.
__device__ __forceinline__ void async_b128(unsigned lds_addr,
                                           const float* gaddr) {
  asm volatile("global_load_async_to_lds_b128 %0, %1, off"
               :
               : "v"(lds_addr), "v"(gaddr)
               : "memory");
}

__device__ __forceinline__ void wait_asynccnt0() {
  asm volatile("s_wait_asynccnt 0x0" ::: "memory");
}

// ---------------------------------------------------------------------------
// TDM: DMA a rows x 128-float 2D f32 tile (row stride 128 elements in memory)
// into LDS with 4-DWORD padding after every 128 DWORDs -> LDS row stride 132.
// ---------------------------------------------------------------------------
__device__ __forceinline__ void tdm_load_tile_128xN(const float* gsrc,
                                                    unsigned ldsByteOff,
                                                    unsigned rows) {
  const unsigned long long ga = (unsigned long long)gsrc;
  v4u g0;
  g0[0] = 1u;                                   // count=1, user descriptor
  g0[1] = ldsByteOff;                           // lds_addr (bytes)
  g0[2] = (unsigned)(ga & 0xFFFFFFFFu);         // global_addr[31:0]
  g0[3] = (unsigned)((ga >> 32) & 0x01FFFFFFu)  // global_addr[56:32]
          | 0x80000000u;                        // type=2 (image)
  v8u g1;
  // data_size=4B (2<<16) | pad_enable (1<<20) | pad_interval=6 -> 128 DWORDs
  // (6<<22) | pad_amount=3 -> 4 DWORDs (3<<25)
  g1[0] = 0x07920000u;
  g1[1] = 128u << 16;                    // tensor_dim0[15:0] = 128
  g1[2] = ((unsigned)L_ & 0xFFFFu) << 16;  // dim0 hi=0 | tensor_dim1 lo = L
  g1[3] = 128u << 16;                    // dim1 hi=0 | tile_dim0 = 128
  g1[4] = rows;                          // tile_dim1 = rows, tile_dim2 = 0
  g1[5] = 128u;                          // tensor_dim0_stride = 128 elements
  g1[6] = 0u;                            // stride0 hi | dim1_stride lo (unused)
  g1[7] = 0u;
  asm volatile("tensor_load_to_lds %0, %1" : : "s"(g0), "s"(g1) : "memory");
}

// ---------------------------------------------------------------------------
// Out = A @ W^T.  A: MxK row-major, W: NxK row-major.
// Block tile 128x128, 4 waves (128 thr, wave32), wave tile 64x64.
// Double-buffered LDS filled by async-to-LDS copies (overlap with WMMA).
// QKV_EPI==1: Out is q-base; k,v derived; RoPE fused; scatter to (B,H,L,D).
// ---------------------------------------------------------------------------
template <int QKV_EPI>
__global__ __launch_bounds__(128) void gemm_wmma_f32(
    const float* __restrict__ A, const float* __restrict__ W,
    float* __restrict__ Out, int M, int N, int K) {
  constexpr int KT  = 16;
  constexpr int LDT = 20;  // 80B row stride: 16B aligned, bank-spread
  __shared__ float lA[2][128 * LDT];
  __shared__ float lB[2][128 * LDT];

  const int tid  = threadIdx.x;
  const int lane = tid & 31;
  const int wave = tid >> 5;
  const int wm   = wave >> 1;  // 0..1
  const int wn   = wave & 1;   // 0..1
  const int lo16 = lane & 15;
  const int hi   = lane >> 4;  // K-pair select for wmma fragments

  const int M0 = blockIdx.y * 128;
  const int N0 = blockIdx.x * 128;

  v8f acc[4][4];
#pragma unroll
  for (int i = 0; i < 4; ++i)
#pragma unroll
    for (int j = 0; j < 4; ++j) acc[i][j] = v8zero();

  const float* aRow = A + (size_t)(M0 + tid) * K;
  const float* wRow = W + (size_t)(N0 + tid) * K;
  const unsigned laOff[2] = {lds_off(&lA[0][tid * LDT]),
                             lds_off(&lA[1][tid * LDT])};
  const unsigned lbOff[2] = {lds_off(&lB[0][tid * LDT]),
                             lds_off(&lB[1][tid * LDT])};

  // issue one 128x16 tile of A and W into LDS buffer `buf` (async)
  auto issue_tiles = [&](int buf, int k0) {
#pragma unroll
    for (int u = 0; u < 4; ++u) {
      async_b128(laOff[buf] + 16u * u, aRow + k0 + 4 * u);
      async_b128(lbOff[buf] + 16u * u, wRow + k0 + 4 * u);
    }
  };

  const int nIter = K / KT;
  issue_tiles(0, 0);
  for (int it = 0; it < nIter; ++it) {
    wait_asynccnt0();   // my copies into buf (it&1) have landed
    __syncthreads();    // everyone's copies landed; buf ((it+1)&1) is free
    if (it + 1 < nIter) issue_tiles((it + 1) & 1, (it + 1) * KT);

    const float* cA = &lA[it & 1][0];
    const float* cB = &lB[it & 1][0];
#pragma unroll
    for (int kb = 0; kb < KT; kb += 4) {
      v2f af[4], bf[4];
#pragma unroll
      for (int i = 0; i < 4; ++i)
        af[i] = *(const v2f*)&cA[(wm * 64 + i * 16 + lo16) * LDT + kb + 2 * hi];
#pragma unroll
      for (int j = 0; j < 4; ++j)
        bf[j] = *(const v2f*)&cB[(wn * 64 + j * 16 + lo16) * LDT + kb + 2 * hi];
#pragma unroll
      for (int i = 0; i < 4; ++i)
#pragma unroll
        for (int j = 0; j < 4; ++j) acc[i][j] = wmma4(af[i], bf[j], acc[i][j]);
    }
  }

  if (QKV_EPI == 0) {
    // plain row-major store
#pragma unroll
    for (int i = 0; i < 4; ++i)
#pragma unroll
      for (int j = 0; j < 4; ++j) {
        const int mB = M0 + wm * 64 + i * 16 + hi * 8;
        const int n  = N0 + wn * 64 + j * 16 + lo16;
#pragma unroll
        for (int r = 0; r < 8; ++r)
          Out[(size_t)(mB + r) * N + n] = acc[i][j][r];
      }
  } else {
    // qkv epilogue: n in [0,6144): part=q/k/v, head, dim. RoPE (head-index
    // rotation, position-independent, per reference) on q and k.
    const size_t QKVSZ = (size_t)B_ * H_ * L_ * D_;
#pragma unroll
    for (int i = 0; i < 4; ++i)
#pragma unroll
      for (int j = 0; j < 4; ++j) {
        const int mB   = M0 + wm * 64 + i * 16 + hi * 8;
        const int n    = N0 + wn * 64 + j * 16 + lo16;
        const int part = n >> 11;
        const int h    = (n >> 7) & 15;
        const int d    = n & 127;
        // theta = h * 10000^{-(d/2)/64} = h * exp(-ln(1e4)/64 * (d/2))
        const float th  = (float)h * __expf(-0.14391157f * (float)(d >> 1));
        const float cth = __cosf(th);
        const float sth = __sinf(th);
        const float sgn = (d & 1) ? 1.0f : -1.0f;
        float* base = Out + (size_t)part * QKVSZ;
#pragma unroll
        for (int r = 0; r < 8; ++r) {
          const int m = mB + r;
          float val   = acc[i][j][r];
          float other = __shfl_xor(val, 1, 32);  // rope pair partner lane
          if (part < 2) {
            // even d: x0*c - x1*s ; odd d: x0*s + x1*c
            val = val * cth + sgn * (other * sth);
          }
          const int b = m >> 11;  // m = b*L + l, L = 2048
          const int l = m & (L_ - 1);
          base[(((size_t)b * H_ + h) * L_ + l) * D_ + d] = val;
        }
      }
  }
}

// ---------------------------------------------------------------------------
// Flash attention: per (b, h, 128-query block). 8 waves x 16 query rows.
// K/V tiles (32x128 f32) DMA'd by the TDM with hardware padding -> LDS row
// stride 132 floats (16B-aligned, bank-conflict-free). QK^T and PV on f32
// WMMA. Row stats replicated across each 16-lane half (matches C/D layout).
// ---------------------------------------------------------------------------
__global__ __launch_bounds__(256) void flash_attn(
    const float* __restrict__ Q, const float* __restrict__ Kk,
    const float* __restrict__ V, float* __restrict__ Y) {
  constexpr int KT  = 32;
  constexpr int LKV = 132;  // produced by TDM pad: +4 DWORDs per 128 DWORDs
  constexpr int LP  = 34;   // 136B row stride: 8B aligned, conflict-free
  __shared__ float lK[KT * LKV];
  __shared__ float lV[KT * LKV];
  __shared__ float lP[8 * 16 * LP];  // per-wave P transpose pads

  const int tid  = threadIdx.x;
  const int lane = tid & 31;
  const int wave = tid >> 5;
  const int lo16 = lane & 15;
  const int hi   = lane >> 4;

  const int q0 = blockIdx.x * 128;
  const int h  = blockIdx.y;
  const int b  = blockIdx.z;
  const size_t bh = ((size_t)b * H_ + h) * (size_t)L_ * D_;
  const float* Qb = Q + bh;
  const float* Kb = Kk + bh;
  const float* Vb = V + bh;

  const unsigned lKOff = lds_off(lK);
  const unsigned lVOff = lds_off(lV);

  const int mQ = wave * 16;  // wave's query rows: [q0+mQ, q0+mQ+16)

  // Q fragments in registers, pre-scaled by 1/sqrt(D)
  v2f aQ[32];
  {
    const float* qrow = Qb + (size_t)(q0 + mQ + lo16) * D_ + 2 * hi;
#pragma unroll
    for (int kk = 0; kk < 32; ++kk) {
      v2f t  = *(const v2f*)(qrow + 4 * kk);
      aQ[kk] = t * 0.08838834764831845f;
    }
  }

  v8f O[8];
#pragma unroll
  for (int j = 0; j < 8; ++j) O[j] = v8zero();
  float mrun[8], lrun[8];
#pragma unroll
  for (int r = 0; r < 8; ++r) { mrun[r] = -3.0e38f; lrun[r] = 0.0f; }

  const int nTiles = q0 / KT + 4;  // causal: keys 0 .. q0+127
  for (int t = 0; t < nTiles; ++t) {
    const int kt0 = t * KT;
    __syncthreads();  // all waves done reading previous K/V tile
    if (wave == 0) {
      tdm_load_tile_128xN(Kb + (size_t)kt0 * D_, lKOff, KT);
      tdm_load_tile_128xN(Vb + (size_t)kt0 * D_, lVOff, KT);
      __builtin_amdgcn_s_wait_tensorcnt(0);
    }
    __syncthreads();  // tile visible to all waves

    if (kt0 <= q0 + mQ + 15) {  // wave-uniform: skip fully-masked tiles
      // S = (Q * scale) @ K^T : 16 x 32
      v8f S[2];
      S[0] = v8zero();
      S[1] = v8zero();
#pragma unroll
      for (int kk = 0; kk < 32; ++kk) {
        v2f b0 = *(const v2f*)&lK[(lo16) * LKV + 4 * kk + 2 * hi];
        v2f b1 = *(const v2f*)&lK[(16 + lo16) * LKV + 4 * kk + 2 * hi];
        S[0]   = wmma4(aQ[kk], b0, S[0]);
        S[1]   = wmma4(aQ[kk], b1, S[1]);
      }
      // causal mask (only tiles overlapping the diagonal)
      if (kt0 + KT - 1 > q0 + mQ) {
#pragma unroll
        for (int j = 0; j < 2; ++j) {
          const int kg = kt0 + j * 16 + lo16;
#pragma unroll
          for (int r = 0; r < 8; ++r) {
            const int qg = q0 + mQ + hi * 8 + r;
            S[j][r]      = (kg > qg) ? -3.0e38f : S[j][r];
          }
        }
      }
      // online softmax update + stage P into per-wave LDS pad
      float* pw = &lP[wave * 16 * LP];
#pragma unroll
      for (int r = 0; r < 8; ++r) {
        float v = fmaxf(S[0][r], S[1][r]);
#pragma unroll
        for (int off = 1; off < 16; off <<= 1)
          v = fmaxf(v, __shfl_xor(v, off, 32));  // reduce within 16-lane half
        const float mnew  = fmaxf(mrun[r], v);
        const float alpha = __expf(mrun[r] - mnew);
        S[0][r]           = __expf(S[0][r] - mnew);
        S[1][r]           = __expf(S[1][r] - mnew);
        float s           = S[0][r] + S[1][r];
#pragma unroll
        for (int off = 1; off < 16; off <<= 1) s += __shfl_xor(s, off, 32);
        lrun[r] = lrun[r] * alpha + s;
        mrun[r] = mnew;
#pragma unroll
        for (int j = 0; j < 8; ++j) O[j][r] *= alpha;
        pw[(r + 8 * hi) * LP + lo16]      = S[0][r];
        pw[(r + 8 * hi) * LP + 16 + lo16] = S[1][r];
      }
      // O += P @ V : (16x32) x (32x128)
#pragma unroll
      for (int kk = 0; kk < 8; ++kk) {
        const int k0v = 4 * kk + 2 * hi;
        v2f aP        = *(const v2f*)&pw[lo16 * LP + k0v];
#pragma unroll
        for (int j = 0; j < 8; ++j) {
          v2f bV;
          bV.x = lV[(k0v) * LKV + j * 16 + lo16];
          bV.y = lV[(k0v + 1) * LKV + j * 16 + lo16];
          O[j] = wmma4(aP, bV, O[j]);
        }
      }
    }
  }

  // normalize and store y in (B, L, C) layout for the projection GEMM
#pragma unroll
  for (int r = 0; r < 8; ++r) {
    const float inv = 1.0f / lrun[r];
    const int qg    = q0 + mQ + hi * 8 + r;
#pragma unroll
    for (int j = 0; j < 8; ++j) {
      const int d = j * 16 + lo16;
      Y[((size_t)b * L_ + qg) * C_ + h * D_ + d] = O[j][r] * inv;
    }
  }
}

// ---------------------------------------------------------------------------
extern "C" void kernel_launch(void* const* d_in, const int* in_sizes, int n_in,
                              void* d_out, int out_size, void* d_ws,
                              size_t ws_size, hipStream_t stream) {
  (void)in_sizes; (void)n_in; (void)out_size; (void)ws_size;
  const float* x      = (const float*)d_in[0];
  const float* w_attn = (const float*)d_in[1];
  const float* w_proj = (const float*)d_in[2];
  float* out = (float*)d_out;
  float* ws  = (float*)d_ws;

  const size_t QKVSZ = (size_t)B_ * H_ * L_ * D_;  // 8 Mi floats per tensor
  float* q = ws;
  float* k = ws + QKVSZ;
  float* v = ws + 2 * QKVSZ;
  float* y = ws + 3 * QKVSZ;  // (B, L, C)

  // 1) qkv = x @ w_attn^T, fused RoPE, scatter to (B,H,L,D)
  dim3 g1(NQKV / 128, (B_ * L_) / 128);
  gemm_wmma_f32<1><<<g1, 128, 0, stream>>>(x, w_attn, q, B_ * L_, NQKV, C_);

  // 2) flash attention
  dim3 g2(L_ / 128, H_, B_);
  flash_attn<<<g2, 256, 0, stream>>>(q, k, v, y);

  // 3) out = y @ w_proj^T
  dim3 g3(C_ / 128, (B_ * L_) / 128);
  gemm_wmma_f32<0><<<g3, 128, 0, stream>>>(y, w_proj, out, B_ * L_, C_, C_);
}